// dcn_4947802325745
// MI455X (gfx1250) — compile-verified
//
#include <hip/hip_runtime.h>

typedef float v2f __attribute__((ext_vector_type(2)));
typedef float v8f __attribute__((ext_vector_type(8)));

#define B_   4
#define C_   64
#define H_   128
#define W_   128
#define HW_  16384
#define COFF 27

#define BSTR 72    // col-matrix stride: 2*BSTR mod 64 == 16 -> halves hit disjoint bank groups
#define ASTR 148   // weight-matrix stride: conflict-free row + half-wave access

// ---------------------------------------------------------------------------
// Kernel 1: conv_offset — 3x3 conv (64 -> 27ch) as implicit GEMM with
// V_WMMA_F32_16X16X4_F32.  M=32 (27 padded), N=64 pixels, K=576.
// K-order within a 16-channel chunk: k = tap*16 + ch  (shift-only decode).
// ---------------------------------------------------------------------------
__global__ __launch_bounds__(256) void conv_offset_kernel(
    const float* __restrict__ c, const float* __restrict__ w_off,
    const float* __restrict__ b_off, float* __restrict__ off_out)
{
  __shared__ __align__(16) float sA[32 * ASTR];    // weights [m][k]
  __shared__ __align__(16) float sB[144 * BSTR];   // im2col  [k][p]

  const int tid = threadIdx.x;
  const int b   = blockIdx.x >> 8;
  const int rem = blockIdx.x & 255;
  const int h   = rem >> 1;
  const int x0  = (rem & 1) * 64;

  const int lane = tid & 31;
  const int wave = tid >> 5;
  const int sub  = lane >> 4;
  const int ln   = lane & 15;
  const int m0   = (wave >> 2) * 16;
  const int n0   = (wave & 3) * 16;

  // lane-constant bases for the unrolled WMMA loop
  const float* aRow = &sA[(m0 + ln) * ASTR + 2 * sub];
  const float* bCol = &sB[144 * sub + n0 + ln];          // (2*sub)*BSTR = 144*sub

  v8f acc = {0.f, 0.f, 0.f, 0.f, 0.f, 0.f, 0.f, 0.f};

  #pragma unroll 1
  for (int cc = 0; cc < 4; ++cc) {                       // 16-ch chunks, Kc=144
    // ---- stage weights A[32][144], rows 27..31 zero ----
    for (int e = tid; e < 32 * 144; e += 256) {
      int m = e / 144, kl = e - m * 144;
      int tap = kl >> 4, chl = kl & 15;
      float v = (m < COFF) ? w_off[(m * C_ + cc * 16 + chl) * 9 + tap] : 0.f;
      sA[m * ASTR + kl] = v;
    }
    // ---- build im2col directly from global (zero-padded halo) ----
    for (int e = tid; e < 144 * 64; e += 256) {          // 36 iters/thread
      int k = e >> 6, p = e & 63;
      int tap = k >> 4, ch = k & 15;
      int r  = tap / 3;
      int gy = h + r - 1;
      int gx = x0 + p + (tap - r * 3) - 1;
      float v = 0.f;
      if (gy >= 0 && gy < H_ && gx >= 0 && gx < W_)
        v = c[((b * C_ + cc * 16 + ch) * H_ + gy) * W_ + gx];
      sB[k * BSTR + p] = v;
    }
    __syncthreads();

    // ---- WMMA: fully unrolled, immediate-offset loads only ----
    #pragma unroll
    for (int kk = 0; kk < 144; kk += 4) {
      v2f a = *(const v2f*)(aRow + kk);
      v2f bv;
      bv.x = bCol[kk * BSTR];
      bv.y = bCol[kk * BSTR + BSTR];
      acc = __builtin_amdgcn_wmma_f32_16x16x4_f32(
          false, a, false, bv, (short)0, acc, false, false);
    }
    __syncthreads();
  }

  #pragma unroll
  for (int j = 0; j < 8; ++j) {
    int m = m0 + j + 8 * sub;
    if (m < COFF)
      off_out[(b * COFF + m) * HW_ + h * W_ + x0 + n0 + ln] = acc[j] + b_off[m];
  }
}

// ---------------------------------------------------------------------------
// Kernel 2: modulated deformable conv = bilinear-gather im2col + f32 WMMA.
// Block: 64 pixels x 64 out channels.  M=64, N=64, K=576 (k = ch*9 + tap so
// the global A fragment stays a contiguous b64).  8 waves x 2 tiles sharing A.
// ---------------------------------------------------------------------------
__global__ __launch_bounds__(256) void dcn_main_kernel(
    const float* __restrict__ x, const float* __restrict__ off,
    const float* __restrict__ w_dcn, const float* __restrict__ b_dcn,
    float* __restrict__ out)
{
  __shared__ __align__(16) float sB[144 * BSTR];  // col chunk [k][p]
  __shared__ int   sIdx[576 * 4];                 // bilinear corner indices
  __shared__ float sW[576 * 4];                   // corner weights * mask

  const int tid  = threadIdx.x;
  const int b    = blockIdx.x >> 8;
  const int tile = blockIdx.x & 255;
  const int p0   = tile * 64;
  const int h    = p0 >> 7;
  const int xw0  = p0 & 127;

  // ---- phase 0: per (pixel, tap) bilinear sampling metadata ----
  for (int it = tid; it < 576; it += 256) {
    int tap = it >> 6;
    int p   = it & 63;
    float dy = off[(b * COFF + 2 * tap    ) * HW_ + p0 + p];
    float dx = off[(b * COFF + 2 * tap + 1) * HW_ + p0 + p];
    float mr = off[(b * COFF + 18 + tap   ) * HW_ + p0 + p];
    float mask = 1.f / (1.f + __expf(-mr));
    float py = (float)h + (float)(tap / 3 - 1) + dy;
    float px = (float)(xw0 + p) + (float)(tap % 3 - 1) + dx;
    float y0f = floorf(py), x0f = floorf(px);
    float wy1 = py - y0f, wx1 = px - x0f;
    float wy0 = 1.f - wy1, wx0 = 1.f - wx1;
    int y0 = (int)y0f, x0i = (int)x0f;
    int y1 = y0 + 1, x1 = x0i + 1;
    float vy0 = (y0  >= 0 && y0  < H_) ? 1.f : 0.f;
    float vy1 = (y1  >= 0 && y1  < H_) ? 1.f : 0.f;
    float vx0 = (x0i >= 0 && x0i < W_) ? 1.f : 0.f;
    float vx1 = (x1  >= 0 && x1  < W_) ? 1.f : 0.f;
    int cy0 = min(max(y0, 0), H_ - 1), cy1 = min(max(y1, 0), H_ - 1);
    int cx0 = min(max(x0i, 0), W_ - 1), cx1 = min(max(x1, 0), W_ - 1);
    sIdx[it * 4 + 0] = cy0 * W_ + cx0;
    sIdx[it * 4 + 1] = cy0 * W_ + cx1;
    sIdx[it * 4 + 2] = cy1 * W_ + cx0;
    sIdx[it * 4 + 3] = cy1 * W_ + cx1;
    sW[it * 4 + 0] = wy0 * wx0 * vy0 * vx0 * mask;
    sW[it * 4 + 1] = wy0 * wx1 * vy0 * vx1 * mask;
    sW[it * 4 + 2] = wy1 * wx0 * vy1 * vx0 * mask;
    sW[it * 4 + 3] = wy1 * wx1 * vy1 * vx1 * mask;
  }
  __syncthreads();

  const int lane = tid & 31;
  const int wave = tid >> 5;
  const int sub  = lane >> 4;
  const int ln   = lane & 15;
  const int mt   = wave >> 1;
  const int ntb  = (wave & 1) * 2;

  // lane-constant bases for the unrolled WMMA loop
  const float* aBase = w_dcn + (size_t)(mt * 16 + ln) * 576 + 2 * sub;
  const float* b0c   = &sB[144 * sub + ntb * 16 + ln];
  const float* b1c   = b0c + 16;

  v8f acc0 = {0.f, 0.f, 0.f, 0.f, 0.f, 0.f, 0.f, 0.f};
  v8f acc1 = {0.f, 0.f, 0.f, 0.f, 0.f, 0.f, 0.f, 0.f};

  #pragma unroll 1
  for (int cc = 0; cc < 4; ++cc) {                 // 16-channel chunks
    // ---- build col chunk sB[ch*9+tap][p] via bilinear gathers ----
    for (int it = tid; it < 576; it += 256) {
      int tap = it >> 6, p = it & 63;
      int i00 = sIdx[it * 4 + 0], i01 = sIdx[it * 4 + 1];
      int i10 = sIdx[it * 4 + 2], i11 = sIdx[it * 4 + 3];
      float w00 = sW[it * 4 + 0], w01 = sW[it * 4 + 1];
      float w10 = sW[it * 4 + 2], w11 = sW[it * 4 + 3];
      const float* xb = x + (size_t)(b * C_ + cc * 16) * HW_;
      #pragma unroll
      for (int ch = 0; ch < 16; ++ch) {
        const float* xp = xb + ch * HW_;
        float v = w00 * xp[i00] + w01 * xp[i01] + w10 * xp[i10] + w11 * xp[i11];
        sB[(ch * 9 + tap) * BSTR + p] = v;
      }
    }
    __syncthreads();

    // ---- WMMA: fully unrolled, immediate-offset loads only ----
    const float* aC = aBase + cc * 144;
    #pragma unroll
    for (int kk = 0; kk < 144; kk += 4) {
      v2f a = *(const v2f*)(aC + kk);
      v2f b0, b1;
      b0.x = b0c[kk * BSTR];
      b0.y = b0c[kk * BSTR + BSTR];
      b1.x = b1c[kk * BSTR];
      b1.y = b1c[kk * BSTR + BSTR];
      acc0 = __builtin_amdgcn_wmma_f32_16x16x4_f32(
          false, a, false, b0, (short)0, acc0, false, false);
      acc1 = __builtin_amdgcn_wmma_f32_16x16x4_f32(
          false, a, false, b1, (short)0, acc1, false, false);
    }
    __syncthreads();
  }

  // ---- epilogue: bias + store ----
  #pragma unroll
  for (int j = 0; j < 8; ++j) {
    int o = mt * 16 + j + 8 * sub;
    float bb = b_dcn[o];
    out[(size_t)(b * C_ + o) * HW_ + p0 + ntb * 16 + ln]       = acc0[j] + bb;
    out[(size_t)(b * C_ + o) * HW_ + p0 + (ntb + 1) * 16 + ln] = acc1[j] + bb;
  }
}

// ---------------------------------------------------------------------------
extern "C" void kernel_launch(void* const* d_in, const int* in_sizes, int n_in,
                              void* d_out, int out_size, void* d_ws, size_t ws_size,
                              hipStream_t stream) {
  const float* x     = (const float*)d_in[0];
  const float* c     = (const float*)d_in[1];
  const float* w_off = (const float*)d_in[2];
  const float* b_off = (const float*)d_in[3];
  const float* w_dcn = (const float*)d_in[4];
  const float* b_dcn = (const float*)d_in[5];
  float* off_out = (float*)d_ws;                // 4*27*16384 floats ~= 7.1 MB
  float* out     = (float*)d_out;

  conv_offset_kernel<<<1024, 256, 0, stream>>>(c, w_off, b_off, off_out);
  dcn_main_kernel<<<1024, 256, 0, stream>>>(x, off_out, w_dcn, b_dcn, out);
}